// GATLayer_40218073760399
// MI455X (gfx1250) — compile-verified
//
#include <hip/hip_runtime.h>
#include <hip/hip_bf16.h>
#include <math.h>

// GAT layer, B=4, N=256, H=4, D=64, IN=128. All fp32.
// Uses V_WMMA_F32_16X16X4_F32 for both GEMMs (full fp32 precision).

typedef float v2f __attribute__((ext_vector_type(2)));
typedef float v8f __attribute__((ext_vector_type(8)));

__device__ __forceinline__ v8f wmma_f32(v2f a, v2f b, v8f c) {
    // 8 args: (neg_a, A, neg_b, B, c_mod, C, reuse_a, reuse_b)
    return __builtin_amdgcn_wmma_f32_16x16x4_f32(
        false, a, false, b, (short)0, c, false, false);
}

// ---------------------------------------------------------------------------
// Kernel 1: Wh[b, 256, 256] = h[b, 256, 128] @ W[128, 256]
// One wave per 16x16 output tile: grid = 4 * 16 * 16 = 1024 blocks of 32.
// ---------------------------------------------------------------------------
__global__ void __launch_bounds__(32) gat_gemm_wh(
    const float* __restrict__ h, const float* __restrict__ W,
    float* __restrict__ Wh) {
    const int tile = blockIdx.x;            // 0..1023
    const int b  = tile >> 8;
    const int mi = (tile >> 4) & 15;
    const int ni = tile & 15;
    const int lane = threadIdx.x;           // 0..31 (wave32)
    const int half = lane >> 4;             // 0 or 1 (K offset 0 / 2)
    const int lm   = lane & 15;             // row (A) / col (B,C)

    const float* Arow = h + (b * 256 + mi * 16 + lm) * 128;  // A row, K-contig
    const float* Bcol = W + ni * 16 + lm;                    // B column base

    v8f c = {};
#pragma unroll
    for (int k = 0; k < 128; k += 4) {
        const int kk = k + 2 * half;
        v2f a = *(const v2f*)(Arow + kk);                 // A[m, kk], A[m, kk+1]
        v2f bm;
        bm.x = Bcol[kk * 256];                            // B[kk,   n]
        bm.y = Bcol[(kk + 1) * 256];                      // B[kk+1, n]
        c = wmma_f32(a, bm, c);
    }

    // C layout: VGPR r, lane -> row = r + 8*half, col = lm
    float* Cp = Wh + (b * 256 + mi * 16) * 256 + ni * 16 + lm;
#pragma unroll
    for (int r = 0; r < 8; ++r)
        Cp[(r + 8 * half) * 256] = c[r];
}

// ---------------------------------------------------------------------------
// Kernel 2: per-row attention scores
//   s1[b,n,hh] = Wh[b,n,hh*64:+64] . a[0:64]
//   s2[b,n,hh] = Wh[b,n,hh*64:+64] . a[64:128]
// One wave per (b,n): 1024 waves.
// ---------------------------------------------------------------------------
__global__ void __launch_bounds__(256) gat_scores(
    const float* __restrict__ Wh, const float* __restrict__ a,
    float* __restrict__ s1, float* __restrict__ s2) {
    const int wave = (blockIdx.x * blockDim.x + threadIdx.x) >> 5;  // 0..1023
    const int lane = threadIdx.x & 31;
    const int b = wave >> 8;
    const int n = wave & 255;
    const float* row = Wh + (b * 256 + n) * 256;
    const int d0 = lane * 2;   // two contiguous elements per lane

#pragma unroll
    for (int hh = 0; hh < 4; ++hh) {
        const float w0 = row[hh * 64 + d0];
        const float w1 = row[hh * 64 + d0 + 1];
        float v1 = w0 * a[d0]      + w1 * a[d0 + 1];
        float v2 = w0 * a[64 + d0] + w1 * a[64 + d0 + 1];
#pragma unroll
        for (int off = 16; off; off >>= 1) {
            v1 += __shfl_xor(v1, off, 32);
            v2 += __shfl_xor(v2, off, 32);
        }
        if (lane == 0) {
            s1[(b * 256 + n) * 4 + hh] = v1;
            s2[(b * 256 + n) * 4 + hh] = v2;
        }
    }
}

// ---------------------------------------------------------------------------
// Kernel 3: masked softmax rows -> att[b,h,i,j]
//   e = leakyrelu(s1[b, h*64+i/4, i%4] + s2[b, (i%4)*64+j/4, j%4])
//   att = softmax_j( adj[b,i,j] ? e : -inf )
// One wave per (b,h,i): 4096 waves, 8 j's per lane.
// ---------------------------------------------------------------------------
__global__ void __launch_bounds__(256) gat_softmax(
    const float* __restrict__ s1, const float* __restrict__ s2,
    const int* __restrict__ adj, float* __restrict__ att) {
    const int wave = (blockIdx.x * blockDim.x + threadIdx.x) >> 5;  // 0..4095
    const int lane = threadIdx.x & 31;
    const int b   = wave >> 10;
    const int rem = wave & 1023;
    const int hh  = rem >> 8;
    const int i   = rem & 255;

    const float s1v = s1[(b * 256 + (hh * 64 + (i >> 2))) * 4 + (i & 3)];
    const int* adjrow = adj + (b * 256 + i) * 256;
    const int r2base = (i & 3) * 64;

    float logit[8];
    float m = -INFINITY;
#pragma unroll
    for (int t = 0; t < 8; ++t) {
        const int j = lane + 32 * t;
        const float s2v = s2[(b * 256 + (r2base + (j >> 2))) * 4 + (j & 3)];
        float e = s1v + s2v;
        e = (e >= 0.0f) ? e : 0.2f * e;          // leaky_relu, slope 0.2
        e = (adjrow[j] != 0) ? e : -INFINITY;    // mask
        logit[t] = e;
        m = fmaxf(m, e);
    }
#pragma unroll
    for (int off = 16; off; off >>= 1) m = fmaxf(m, __shfl_xor(m, off, 32));

    float sum = 0.0f;
#pragma unroll
    for (int t = 0; t < 8; ++t) {
        const float p = __expf(logit[t] - m);
        logit[t] = p;
        sum += p;
    }
#pragma unroll
    for (int off = 16; off; off >>= 1) sum += __shfl_xor(sum, off, 32);

    const float inv = 1.0f / sum;
    float* outrow = att + ((b * 4 + hh) * 256 + i) * 256;
#pragma unroll
    for (int t = 0; t < 8; ++t) outrow[lane + 32 * t] = logit[t] * inv;
}

// ---------------------------------------------------------------------------
// Kernel 4: out[b, i, hh*64 + d] = sum_j att[b,hh,i,j] * Wh[b, j, hh*64 + d]
// Per (b,hh): 256x64 = att(256x256) @ Wh-slice(256x64).
// One wave per 16x16 tile: 4*4*16*4 = 1024 blocks of 32.
// ---------------------------------------------------------------------------
__global__ void __launch_bounds__(32) gat_gemm_out(
    const float* __restrict__ att, const float* __restrict__ Wh,
    float* __restrict__ out) {
    const int tile = blockIdx.x;            // ((b*4+hh)*16 + it)*4 + dt
    const int dt = tile & 3;
    const int it = (tile >> 2) & 15;
    const int hh = (tile >> 6) & 3;
    const int b  = tile >> 8;
    const int lane = threadIdx.x;
    const int half = lane >> 4;
    const int lm   = lane & 15;

    const float* Arow = att + ((b * 4 + hh) * 256 + it * 16 + lm) * 256;
    const float* Bcol = Wh + b * 256 * 256 + hh * 64 + dt * 16 + lm;

    v8f c = {};
#pragma unroll 8
    for (int k = 0; k < 256; k += 4) {
        const int kk = k + 2 * half;
        v2f a = *(const v2f*)(Arow + kk);
        v2f bm;
        bm.x = Bcol[kk * 256];
        bm.y = Bcol[(kk + 1) * 256];
        c = wmma_f32(a, bm, c);
    }

    float* Cp = out + (b * 256 + it * 16) * 256 + hh * 64 + dt * 16 + lm;
#pragma unroll
    for (int r = 0; r < 8; ++r)
        Cp[(r + 8 * half) * 256] = c[r];
}

// ---------------------------------------------------------------------------
extern "C" void kernel_launch(void* const* d_in, const int* in_sizes, int n_in,
                              void* d_out, int out_size, void* d_ws, size_t ws_size,
                              hipStream_t stream) {
    const float* h   = (const float*)d_in[0];   // [4,256,128]
    const int*   adj = (const int*)d_in[1];     // [4,256,256]
    const float* W   = (const float*)d_in[2];   // [128,256]
    const float* a   = (const float*)d_in[3];   // [128]
    float* out = (float*)d_out;                 // [4,256,256]

    char* ws = (char*)d_ws;
    float* Wh  = (float*)(ws);                                   // 262144 floats
    float* s1  = (float*)(ws + (size_t)262144 * 4);              //   4096 floats
    float* s2  = (float*)(ws + (size_t)266240 * 4);              //   4096 floats
    float* att = (float*)(ws + (size_t)270336 * 4);              // 1048576 floats

    gat_gemm_wh <<<1024, 32,  0, stream>>>(h, W, Wh);
    gat_scores  <<<128,  256, 0, stream>>>(Wh, a, s1, s2);
    gat_softmax <<<512,  256, 0, stream>>>(s1, s2, adj, att);
    gat_gemm_out<<<1024, 32,  0, stream>>>(att, Wh, out);
}